// EvolveGCNOModel_50483045597459
// MI455X (gfx1250) — compile-verified
//
#include <hip/hip_runtime.h>
#include <hip/hip_bf16.h>

// ---------------------------------------------------------------------------
// EvolveGCNO on MI455X (gfx1250, wave32).
// Roofline: edge gather/scatter (~0.8GB random) dominates at 23.3 TB/s HBM;
// GEMMs (2 x 0.8 GFLOP) are negligible -> keep fp32, run them on the WMMA
// pipe (V_WMMA_F32_16X16X4_F32), use hardware fp32 atomics for the scatter.
// ---------------------------------------------------------------------------

#define NN 100000
#define NE 1600000
#define FD 64

typedef __attribute__((ext_vector_type(2))) float v2f;
typedef __attribute__((ext_vector_type(8))) float v8f;

// --- K1: GRU weight evolution (one workgroup; tiny) ------------------------
__global__ __launch_bounds__(256) void k_gru_evolve(
    const float* __restrict__ W0, const float* __restrict__ Wi,
    const float* __restrict__ Wh, const float* __restrict__ bi,
    const float* __restrict__ bh, float* __restrict__ W)
{
    __shared__ float sW0[FD * FD];
    for (int i = threadIdx.x; i < FD * FD; i += 256) sW0[i] = W0[i];
    __syncthreads();
    for (int p = threadIdx.x; p < FD * FD; p += 256) {
        const int i = p >> 6, j = p & 63;
        const float* w0row = &sW0[i * FD];
        float ir = bi[j], iz = bi[FD + j], in_ = bi[2 * FD + j];
        float hr = bh[j], hz = bh[FD + j], hn  = bh[2 * FD + j];
        for (int k = 0; k < FD; ++k) {
            const float a = w0row[k];
            ir  += a * Wi[(j)          * FD + k];
            iz  += a * Wi[(FD + j)     * FD + k];
            in_ += a * Wi[(2 * FD + j) * FD + k];
            hr  += a * Wh[(j)          * FD + k];
            hz  += a * Wh[(FD + j)     * FD + k];
            hn  += a * Wh[(2 * FD + j) * FD + k];
        }
        const float r = 1.f / (1.f + __expf(-(ir + hr)));
        const float z = 1.f / (1.f + __expf(-(iz + hz)));
        const float n = tanhf(in_ + r * hn);
        W[p] = (1.f - z) * n + z * sW0[p];
    }
}

// --- K2: deg init (self-loop weight 1) -------------------------------------
__global__ void k_deg_init(float* __restrict__ deg)
{
    const int i = blockIdx.x * blockDim.x + threadIdx.x;
    if (i < NN) deg[i] = 1.0f;
}

// --- K3: deg[dst] += w  (hardware fp32 atomic) -----------------------------
__global__ void k_deg_edges(const long long* __restrict__ ei,
                            const float* __restrict__ ew,
                            float* __restrict__ deg)
{
    const int e = blockIdx.x * blockDim.x + threadIdx.x;
    if (e < NE) {
        const int d = (int)ei[NE + e];
        unsafeAtomicAdd(&deg[d], ew[e]);
    }
}

// --- K4: dinv = rsqrt(deg) -------------------------------------------------
__global__ void k_dinv(const float* __restrict__ deg, float* __restrict__ dinv)
{
    const int i = blockIdx.x * blockDim.x + threadIdx.x;
    if (i < NN) {
        const float d = deg[i];
        dinv[i] = d > 0.f ? rsqrtf(d) : 0.f;
    }
}

// --- K5: xw = x @ W via V_WMMA_F32_16X16X4_F32 -----------------------------
// One wave per 16-row slab (6250 slabs). A preloaded once, 4 col tiles.
__global__ __launch_bounds__(256) void k_xw_wmma(
    const float* __restrict__ x, const float* __restrict__ W,
    float* __restrict__ xw)
{
    const int wave = (blockIdx.x * blockDim.x + threadIdx.x) >> 5;
    if (wave >= NN / 16) return;                 // uniform per wave: EXEC stays all-1
    const int lane = threadIdx.x & 31;
    const int half = lane >> 4, lp = lane & 15;
    const int row0 = wave * 16;

    // A[16x4] chunk kk: lane<16 holds {K=4kk,4kk+1}; lane>=16 holds {K=4kk+2,4kk+3}
    v2f a[16];
    const float* arow = x + (size_t)(row0 + lp) * FD + 2 * half;
#pragma unroll
    for (int kk = 0; kk < 16; ++kk) {
        a[kk].x = arow[kk * 4 + 0];
        a[kk].y = arow[kk * 4 + 1];
    }
#pragma unroll
    for (int ct = 0; ct < 4; ++ct) {
        const int col = ct * 16 + lp;
        v8f c = {0.f, 0.f, 0.f, 0.f, 0.f, 0.f, 0.f, 0.f};
#pragma unroll
        for (int kk = 0; kk < 16; ++kk) {
            const int kr = kk * 4 + 2 * half;    // B[4x16]: mirrors A layout
            v2f b;
            b.x = W[kr * FD + col];
            b.y = W[(kr + 1) * FD + col];
            c = __builtin_amdgcn_wmma_f32_16x16x4_f32(
                    false, a[kk], false, b, (short)0, c, false, false);
        }
#pragma unroll
        for (int r = 0; r < 8; ++r)              // C: M = r + 8*half, N = col
            xw[(size_t)(row0 + r + 8 * half) * FD + col] = c[r];
    }
}

// --- K6: h init with self-loop term: h = xw * dinv^2 -----------------------
__global__ void k_h_init(const float* __restrict__ xw,
                         const float* __restrict__ dinv,
                         float* __restrict__ h)
{
    const int t = blockIdx.x * blockDim.x + threadIdx.x;
    if (t < NN * FD) {
        const float di = dinv[t >> 6];
        h[t] = xw[t] * di * di;
    }
}

// --- K7: edge aggregation, h[dst] += xw[src] * (dinv_s * w * dinv_d) -------
__global__ void k_edge_agg(const long long* __restrict__ ei,
                           const float* __restrict__ ew,
                           const float* __restrict__ dinv,
                           const float* __restrict__ xw,
                           float* __restrict__ h)
{
    const int t = blockIdx.x * blockDim.x + threadIdx.x;  // NE*64 < 2^31
    const int e = t >> 6;
    if (e >= NE) return;
    const int f = t & 63;
    const int s = (int)ei[e];
    const int d = (int)ei[NE + e];
    const float coef = dinv[s] * ew[e] * dinv[d];
    unsafeAtomicAdd(&h[(size_t)d * FD + f], xw[(size_t)s * FD + f] * coef);
}

// --- K8: fused head: out = relu(h @ projW^T + pb) . linW + lb --------------
__global__ __launch_bounds__(256) void k_head_wmma(
    const float* __restrict__ h, const float* __restrict__ projW,
    const float* __restrict__ projB, const float* __restrict__ linW,
    const float* __restrict__ linB, float* __restrict__ out)
{
    const int wave = (blockIdx.x * blockDim.x + threadIdx.x) >> 5;
    if (wave >= NN / 16) return;
    const int lane = threadIdx.x & 31;
    const int half = lane >> 4, lp = lane & 15;
    const int row0 = wave * 16;

    v2f a[16];
    const float* arow = h + (size_t)(row0 + lp) * FD + 2 * half;
#pragma unroll
    for (int kk = 0; kk < 16; ++kk) {
        a[kk].x = arow[kk * 4 + 0];
        a[kk].y = arow[kk * 4 + 1];
    }

    float acc[8] = {0.f, 0.f, 0.f, 0.f, 0.f, 0.f, 0.f, 0.f};
#pragma unroll
    for (int ct = 0; ct < 4; ++ct) {
        const int j = ct * 16 + lp;              // hidden-unit index
        v8f c = {0.f, 0.f, 0.f, 0.f, 0.f, 0.f, 0.f, 0.f};
#pragma unroll
        for (int kk = 0; kk < 16; ++kk) {
            const int kr = kk * 4 + 2 * half;
            v2f b;                               // B[k][j] = projW[j*64 + k] (contig pair)
            b.x = projW[(size_t)j * FD + kr];
            b.y = projW[(size_t)j * FD + kr + 1];
            c = __builtin_amdgcn_wmma_f32_16x16x4_f32(
                    false, a[kk], false, b, (short)0, c, false, false);
        }
        const float pb = projB[j], lw = linW[j];
#pragma unroll
        for (int r = 0; r < 8; ++r) {
            float v = c[r] + pb;
            v = v > 0.f ? v : 0.f;               // ReLU
            acc[r] += v * lw;                    // head dot, partial over cols
        }
    }
    // reduce across the 16 lanes of each half (masks < 16 stay within half)
#pragma unroll
    for (int m = 1; m < 16; m <<= 1)
#pragma unroll
        for (int r = 0; r < 8; ++r)
            acc[r] += __shfl_xor(acc[r], m, 32);
    if (lp == 0) {
        const float lb = linB[0];
#pragma unroll
        for (int r = 0; r < 8; ++r)
            out[row0 + r + 8 * half] = acc[r] + lb;
    }
}

// ---------------------------------------------------------------------------
extern "C" void kernel_launch(void* const* d_in, const int* in_sizes, int n_in,
                              void* d_out, int out_size, void* d_ws, size_t ws_size,
                              hipStream_t stream)
{
    const float*     x   = (const float*)d_in[0];
    const long long* ei  = (const long long*)d_in[1];   // int64 [2, NE]
    const float*     ew  = (const float*)d_in[2];
    const float*     W0  = (const float*)d_in[3];
    const float*     gWi = (const float*)d_in[4];
    const float*     gWh = (const float*)d_in[5];
    const float*     gbi = (const float*)d_in[6];
    const float*     gbh = (const float*)d_in[7];
    const float*     pW  = (const float*)d_in[8];
    const float*     pb  = (const float*)d_in[9];
    const float*     lW  = (const float*)d_in[10];
    const float*     lb  = (const float*)d_in[11];
    float* outp = (float*)d_out;

    // workspace layout (floats): W | deg | dinv | xw | h   (~52 MB total)
    float* ws   = (float*)d_ws;
    float* W    = ws;                       // 4096
    float* deg  = W + 4096;                 // NN
    float* dinv = deg + NN;                 // NN
    float* xw   = dinv + NN;                // NN*FD
    float* h    = xw + (size_t)NN * FD;     // NN*FD

    k_gru_evolve<<<1, 256, 0, stream>>>(W0, gWi, gWh, gbi, gbh, W);
    k_deg_init<<<(NN + 255) / 256, 256, 0, stream>>>(deg);
    k_deg_edges<<<(NE + 255) / 256, 256, 0, stream>>>(ei, ew, deg);
    k_dinv<<<(NN + 255) / 256, 256, 0, stream>>>(deg, dinv);

    const int slabs = NN / 16;                     // 6250, exact
    const int wblk  = (slabs + 7) / 8;             // 8 waves (256 thr) per block
    k_xw_wmma<<<wblk, 256, 0, stream>>>(x, W, xw);

    k_h_init<<<(NN * FD + 255) / 256, 256, 0, stream>>>(xw, dinv, h);
    k_edge_agg<<<(NE * 64) / 256, 256, 0, stream>>>(ei, ew, dinv, xw, h);

    k_head_wmma<<<wblk, 256, 0, stream>>>(h, pW, pb, lW, lb, outp);
}